// GCNConv_36249523978360
// MI455X (gfx1250) — compile-verified
//
#include <hip/hip_runtime.h>

typedef __attribute__((ext_vector_type(2))) float v2f;
typedef __attribute__((ext_vector_type(4))) float v4f;
typedef __attribute__((ext_vector_type(8))) float v8f;

#define N_NODES 100000
#define IN_F    256
#define OUT_F   128
#define N_EDGES 1600000

// -----------------------------------------------------------------------------
// Kernel 1: h = x @ W using V_WMMA_F32_16X16X4_F32 (fp32 exact, matches ref).
// One wave32 computes one 16x16 tile of h. blockDim=256 -> 8 waves cover the
// full OUT_F=128 strip for one 16-row block of nodes. grid.x = 100000/16 = 6250
// (exact, no bounds checks).
//
// VGPR layouts (ISA 7.12.2, wave32):
//   A (16x4 f32, 2 VGPRs): lanes 0-15 hold M=lane, {K=0,K=1}; lanes 16-31 hold
//     M=lane-16, {K=2,K=3}  -> contiguous K pair => one float2 load per lane.
//   B (4x16 f32, 2 VGPRs): mirrored: lanes 0-15 N=lane {K=0,K=1}, lanes 16-31
//     N=lane-16 {K=2,K=3}   -> two stride-OUT_F scalar loads.
//   C/D (16x16 f32, 8 VGPRs): VGPR v: lanes 0-15 -> M=v, N=lane;
//     lanes 16-31 -> M=v+8, N=lane-16.
// -----------------------------------------------------------------------------
__global__ __launch_bounds__(256) void gcn_gemm_wmma(
    const float* __restrict__ x,      // [N_NODES, IN_F]
    const float* __restrict__ w,      // [IN_F, OUT_F]
    float* __restrict__ h)            // [N_NODES, OUT_F]
{
  const int wave = threadIdx.x >> 5;   // 0..7 : which 16-col tile of OUT_F
  const int lane = threadIdx.x & 31;
  const int half = lane >> 4;          // 0 or 1
  const int idx  = lane & 15;
  const int row0 = blockIdx.x * 16;
  const int col0 = wave * 16;

  const float* __restrict__ xrow = x + (size_t)(row0 + idx) * IN_F;
  const float* __restrict__ wcol = w + col0 + idx;

  v8f acc = {};
#pragma unroll 8
  for (int kk = 0; kk < IN_F; kk += 4) {
    const int ka = kk + 2 * half;                 // even -> 8B aligned
    v2f a = *(const v2f*)(xrow + ka);             // A[M=idx][ka], [ka+1]
    v2f b;
    b.x = wcol[(size_t)ka * OUT_F];               // B[ka][N=idx]
    b.y = wcol[(size_t)(ka + 1) * OUT_F];         // B[ka+1][N=idx]
    // 8 args: (neg_a, A, neg_b, B, c_mod, C, reuse_a, reuse_b)
    acc = __builtin_amdgcn_wmma_f32_16x16x4_f32(
        false, a, false, b, (short)0, acc, false, false);
  }

  // D layout: VGPR v -> row (v + 8*half), col idx within tile
  float* __restrict__ hp = h + (size_t)(row0 + 8 * half) * OUT_F + col0 + idx;
#pragma unroll
  for (int v = 0; v < 8; ++v) {
    hp[(size_t)v * OUT_F] = acc[v];
  }
}

// -----------------------------------------------------------------------------
// Kernel 2: zero d_out (harness poisons it). float4 stores, exact grid.
// -----------------------------------------------------------------------------
__global__ __launch_bounds__(256) void gcn_zero(v4f* __restrict__ out) {
  const int i = blockIdx.x * blockDim.x + threadIdx.x;   // 0 .. 3.2M-1
  out[i] = (v4f){0.f, 0.f, 0.f, 0.f};
}

// -----------------------------------------------------------------------------
// Kernel 3: SpMM scatter-add. One edge per wave32; lane L owns features
// [4L, 4L+3] as a float4. h (51.2 MB) and out (51.2 MB) both fit in the 192 MB
// L2, so the gather + atomic streams are L2-resident. Relaxed agent-scope fp32
// atomic adds (no return -> STOREcnt path, executed at L2).
// -----------------------------------------------------------------------------
__global__ __launch_bounds__(256) void gcn_spmm(
    const int*   __restrict__ erow,
    const int*   __restrict__ ecol,
    const float* __restrict__ eval,
    const float* __restrict__ h,
    float* __restrict__ out)
{
  const int edge = (blockIdx.x * 256 + threadIdx.x) >> 5;  // wave id == edge id
  const int lane = threadIdx.x & 31;
  if (edge >= N_EDGES) return;

  const int   r = erow[edge];     // uniform within wave -> coalesces
  const int   c = ecol[edge];
  const float v = eval[edge];

  const v4f hv  = *(const v4f*)(h + (size_t)c * OUT_F + 4 * lane);
  const v4f msg = hv * v;

  float* __restrict__ o = out + (size_t)r * OUT_F + 4 * lane;
  __hip_atomic_fetch_add(o + 0, msg.x, __ATOMIC_RELAXED, __HIP_MEMORY_SCOPE_AGENT);
  __hip_atomic_fetch_add(o + 1, msg.y, __ATOMIC_RELAXED, __HIP_MEMORY_SCOPE_AGENT);
  __hip_atomic_fetch_add(o + 2, msg.z, __ATOMIC_RELAXED, __HIP_MEMORY_SCOPE_AGENT);
  __hip_atomic_fetch_add(o + 3, msg.w, __ATOMIC_RELAXED, __HIP_MEMORY_SCOPE_AGENT);
}

extern "C" void kernel_launch(void* const* d_in, const int* in_sizes, int n_in,
                              void* d_out, int out_size, void* d_ws, size_t ws_size,
                              hipStream_t stream) {
  (void)in_sizes; (void)n_in; (void)out_size; (void)ws_size;

  const float* x    = (const float*)d_in[0];   // [N_NODES, IN_F]
  const float* w    = (const float*)d_in[1];   // [IN_F, OUT_F]
  const int*   erow = (const int*)  d_in[2];   // [N_EDGES]
  const int*   ecol = (const int*)  d_in[3];   // [N_EDGES]
  const float* ev   = (const float*)d_in[4];   // [N_EDGES]
  float* out = (float*)d_out;                  // [N_NODES, OUT_F]
  float* h   = (float*)d_ws;                   // [N_NODES, OUT_F] scratch (51.2 MB)

  // h = x @ W   (WMMA fp32)
  gcn_gemm_wmma<<<N_NODES / 16, 256, 0, stream>>>(x, w, h);

  // out = 0
  gcn_zero<<<(N_NODES * OUT_F / 4) / 256, 256, 0, stream>>>((v4f*)out);

  // out[r] += v * h[c]  (one edge per wave32)
  const int spmm_blocks = (N_EDGES * 32) / 256;  // 200000, exact
  gcn_spmm<<<spmm_blocks, 256, 0, stream>>>(erow, ecol, ev, h, out);
}